// ABCNN2Attention_77996606095982
// MI455X (gfx1250) — compile-verified
//
#include <hip/hip_runtime.h>

#define BATCH   128
#define MTOT    259       // MAX_LENGTH + WIDTH - 1
#define MAXLEN  256
#define WIDTH   4
#define DDIM    512
#define NTILE   17        // ceil(259/16)
#define LDSPITCH 516      // 512 + 4 pad -> conflict-free A-frag reads

typedef __attribute__((ext_vector_type(2))) float v2f;
typedef __attribute__((ext_vector_type(4))) float v4f;
typedef __attribute__((ext_vector_type(8))) float v8f;

__device__ __forceinline__ int imin_i(int a, int b) { return a < b ? a : b; }

// ---------------------------------------------------------------------------
// Kernel 1: per-row squared norms of a and b, plus zero-fill of the w_a / w_b
// accumulators (done every call -> deterministic under graph replay).
// One block per batch, 8 waves, one row per wave per iteration.
// ---------------------------------------------------------------------------
__global__ __launch_bounds__(256)
void norms_zero_kernel(const float* __restrict__ a, const float* __restrict__ b,
                       float* __restrict__ na, float* __restrict__ nb,
                       float* __restrict__ wa, float* __restrict__ wb) {
    const int batch = blockIdx.x;
    const int tid   = threadIdx.x;
    const int wave  = tid >> 5;
    const int lane  = tid & 31;

    for (int m = tid; m < MTOT; m += 256) {
        wa[batch * MTOT + m] = 0.0f;
        wb[batch * MTOT + m] = 0.0f;
    }

    const float* abase = a + (size_t)batch * MTOT * DDIM;
    const float* bbase = b + (size_t)batch * MTOT * DDIM;

    for (int row = wave; row < MTOT; row += 8) {
        const v4f* ar = (const v4f*)(abase + (size_t)row * DDIM);
        const v4f* br = (const v4f*)(bbase + (size_t)row * DDIM);
        float sa = 0.0f, sb = 0.0f;
#pragma unroll
        for (int it = 0; it < DDIM / 128; ++it) {       // 4 iterations
            v4f va = ar[lane + 32 * it];
            v4f vb = br[lane + 32 * it];
            sa += va.x * va.x + va.y * va.y + va.z * va.z + va.w * va.w;
            sb += vb.x * vb.x + vb.y * vb.y + vb.z * vb.z + vb.w * vb.w;
        }
#pragma unroll
        for (int m = 16; m >= 1; m >>= 1) {
            sa += __shfl_xor(sa, m, 32);
            sb += __shfl_xor(sb, m, 32);
        }
        if (lane == 0) {
            na[batch * MTOT + row] = sa;
            nb[batch * MTOT + row] = sb;
        }
    }
}

// ---------------------------------------------------------------------------
// Kernel 2: Gram matrix via V_WMMA_F32_16X16X4_F32, fused with the
// dist -> 1/(1+d) transform and row/col marginal sums of A.
// Grid: (i_tile, batch). 8 waves per block; wave w handles j-tiles w, w+8, ...
//
// F32 WMMA fragment layouts (ISA 7.12.2):
//   A (16x4): lane L holds row M=(L&15); VGPR0/1 = K = k0+2*(L>>4) .. +1
//   B (4x16): lane L holds col N=(L&15); VGPR0/1 = K = k0+2*(L>>4) .. +1
//   C/D 16x16: VGPR r, lane L -> element (M = r + 8*(L>>4), N = L&15)
// Both A and B fragments are contiguous per-lane float2 loads.
// ---------------------------------------------------------------------------
__global__ __launch_bounds__(256)
void attn_sums_kernel(const float* __restrict__ a, const float* __restrict__ b,
                      const float* __restrict__ na, const float* __restrict__ nb,
                      float* __restrict__ wa, float* __restrict__ wb) {
    __shared__ float lds_a[16 * LDSPITCH];              // 33 KB of 320 KB

    const int itile = blockIdx.x;
    const int batch = blockIdx.y;
    const int i0    = itile * 16;
    const int tid   = threadIdx.x;
    const int wave  = tid >> 5;
    const int lane  = tid & 31;
    const int lo    = lane & 15;
    const int hi    = lane >> 4;

    const float* abase = a + (size_t)batch * MTOT * DDIM;
    const float* bbase = b + (size_t)batch * MTOT * DDIM;

    // Stage the 16 x 512 a-strip into LDS (rows clamped at the edge tile).
    for (int c = tid; c < 16 * (DDIM / 4); c += 256) {
        int row  = c >> 7;                // 128 float4 per row
        int col4 = c & 127;
        int grow = imin_i(i0 + row, MTOT - 1);
        v4f v = *(const v4f*)(abase + (size_t)grow * DDIM + col4 * 4);
        *(v4f*)(&lds_a[row * LDSPITCH + col4 * 4]) = v;
    }
    __syncthreads();

    // This lane's 8 accumulator rows: i = i0 + r + 8*hi
    float nav[8];
#pragma unroll
    for (int r = 0; r < 8; ++r)
        nav[r] = na[batch * MTOT + imin_i(i0 + r + 8 * hi, MTOT - 1)];

    const float* aptr = &lds_a[lo * LDSPITCH + 2 * hi];

    float rs[8];
#pragma unroll
    for (int r = 0; r < 8; ++r) rs[r] = 0.0f;

    for (int jt = wave; jt < NTILE; jt += 8) {
        const int j0 = jt * 16;
        const int j  = j0 + lo;
        const int jc = imin_i(j, MTOT - 1);
        const bool jvalid = (j < MTOT);
        const float nbv = nb[batch * MTOT + jc];
        const float* bptr = bbase + (size_t)jc * DDIM + 2 * hi;

        v8f acc = {};
#pragma unroll 8
        for (int k0 = 0; k0 < DDIM; k0 += 4) {
            v2f av = *(const v2f*)(aptr + k0);          // ds_load_b64
            v2f bv = *(const v2f*)(bptr + k0);          // global_load_b64
            acc = __builtin_amdgcn_wmma_f32_16x16x4_f32(
                false, av, false, bv, (short)0, acc, false, false);
        }

        // acc[r] = dot(a_i, b_j) with i = i0 + r + 8*hi, j = j0 + lo
        float cs = 0.0f;
#pragma unroll
        for (int r = 0; r < 8; ++r) {
            int i = i0 + r + 8 * hi;
            float sq = nav[r] + nbv - 2.0f * acc[r];
            sq = sq > 0.0f ? sq : 0.0f;
            float Aij = 1.0f / (1.0f + __builtin_sqrtf(sq));
            if (i >= MTOT || !jvalid) Aij = 0.0f;
            rs[r] += Aij;                                // partial row sum
            cs    += Aij;                                // partial col sum
        }
        cs += __shfl_xor(cs, 16, 32);                    // combine M and M+8 halves
        if (hi == 0 && jvalid)
            atomicAdd(&wa[batch * MTOT + j], cs);        // column sums of A
    }

    // Row sums: reduce over the 16 lanes of each half (j direction).
#pragma unroll
    for (int r = 0; r < 8; ++r) {
        float v = rs[r];
        v += __shfl_xor(v, 1, 32);
        v += __shfl_xor(v, 2, 32);
        v += __shfl_xor(v, 4, 32);
        v += __shfl_xor(v, 8, 32);
        int i = i0 + r + 8 * hi;
        if (lo == 0 && i < MTOT)
            atomicAdd(&wb[batch * MTOT + i], v);         // row sums of A
    }
}

// ---------------------------------------------------------------------------
// Kernel 3: w1[b,t,:] = sum_{k<4} wa[b,t+k]*a[b,t+k,:]  (and w2 with wb, b).
// Grid: (t, batch); 128 threads, one float4 column each.
// ---------------------------------------------------------------------------
__global__ __launch_bounds__(128)
void output_kernel(const float* __restrict__ a, const float* __restrict__ b,
                   const float* __restrict__ wa, const float* __restrict__ wb,
                   float* __restrict__ out1, float* __restrict__ out2) {
    const int t     = blockIdx.x;
    const int batch = blockIdx.y;
    const int c     = threadIdx.x;                       // float4 column index

    const float* abase = a + (size_t)batch * MTOT * DDIM;
    const float* bbase = b + (size_t)batch * MTOT * DDIM;

    v4f s1 = {}, s2 = {};
#pragma unroll
    for (int k = 0; k < WIDTH; ++k) {
        int m = t + k;
        float fa = wa[batch * MTOT + m];
        float fb = wb[batch * MTOT + m];
        v4f va = *(const v4f*)(abase + (size_t)m * DDIM + c * 4);
        v4f vb = *(const v4f*)(bbase + (size_t)m * DDIM + c * 4);
        s1 += fa * va;
        s2 += fb * vb;
    }
    size_t o = ((size_t)batch * MAXLEN + t) * (size_t)DDIM + c * 4;
    *(v4f*)(out1 + o) = s1;
    *(v4f*)(out2 + o) = s2;
}

// ---------------------------------------------------------------------------
extern "C" void kernel_launch(void* const* d_in, const int* in_sizes, int n_in,
                              void* d_out, int out_size, void* d_ws, size_t ws_size,
                              hipStream_t stream) {
    const float* x1 = (const float*)d_in[0];   // (128, 1, 259, 512) fp32
    const float* x2 = (const float*)d_in[1];   // (128, 1, 259, 512) fp32
    float* out = (float*)d_out;                // w1 then w2, each (128,1,256,512)
    float* ws  = (float*)d_ws;

    float* na = ws;                             // 128*259
    float* nb = na + BATCH * MTOT;
    float* wa = nb + BATCH * MTOT;
    float* wb = wa + BATCH * MTOT;              // total ~530 KB of workspace

    norms_zero_kernel<<<dim3(BATCH), dim3(256), 0, stream>>>(x1, x2, na, nb, wa, wb);

    attn_sums_kernel<<<dim3(NTILE, BATCH), dim3(256), 0, stream>>>(x1, x2, na, nb, wa, wb);

    float* out1 = out;
    float* out2 = out + (size_t)BATCH * MAXLEN * DDIM;
    output_kernel<<<dim3(MAXLEN, BATCH), dim3(128), 0, stream>>>(x1, x2, wa, wb, out1, out2);
}